// MambaTemporalStream_27041114095825
// MI455X (gfx1250) — compile-verified
//
#include <hip/hip_runtime.h>

#define D_MODEL 512
#define D_INNER 1024
#define D_STATE 16
#define DT_RANK 32
#define SEQ     2048
#define B_SZ    8
#define M_ROWS  (B_SZ*SEQ)   // 16384

typedef float v2f __attribute__((ext_vector_type(2)));
typedef float v8f __attribute__((ext_vector_type(8)));

// ---------------------------------------------------------------- LayerNorm
// one wave32 per 512-element row; 8 waves / block
__global__ __launch_bounds__(256)
void ln_kernel(const float* __restrict__ x, const float* __restrict__ w,
               const float* __restrict__ b, float* __restrict__ o)
{
    const int lane = threadIdx.x & 31;
    const int wv   = threadIdx.x >> 5;
    const size_t row = (size_t)blockIdx.x * 8 + wv;
    const float* xr = x + row * D_MODEL;
    float v[16];
    float s = 0.f;
#pragma unroll
    for (int i = 0; i < 16; ++i) { v[i] = xr[lane + i*32]; s += v[i]; }
#pragma unroll
    for (int m = 16; m >= 1; m >>= 1) s += __shfl_xor(s, m, 32);
    const float mu = s * (1.f/512.f);
    float q = 0.f;
#pragma unroll
    for (int i = 0; i < 16; ++i) { float d = v[i]-mu; q += d*d; }
#pragma unroll
    for (int m = 16; m >= 1; m >>= 1) q += __shfl_xor(q, m, 32);
    const float rs = rsqrtf(q * (1.f/512.f) + 1e-5f);
    float* orow = o + row * D_MODEL;
#pragma unroll
    for (int i = 0; i < 16; ++i) {
        const int c = lane + i*32;
        orow[c] = (v[i]-mu) * rs * w[c] + b[c];
    }
}

// ---------------------------------------------------------------- WMMA GEMM
// C[M,N] = A[M,K] * W[N,K]^T , f32 WMMA 16x16x4.
// block = 128 thr (4 waves); block tile = 16 (M) x 256 (N); wave tile 16x64.
// Double-buffered LDS A-tile (1 barrier / 32-K chunk), B-frags batch-loaded
// into registers so WMMAs issue back-to-back.
// EPI: 0 = store, 1 = softplus(acc + bias[n]), 2 = acc + residual[m,n]
#define KC 32
#define LDSA_STRIDE 34

__device__ __forceinline__ v8f wmma4(v2f a, v2f b, v8f c) {
    return __builtin_amdgcn_wmma_f32_16x16x4_f32(
        false, a, false, b, (short)0, c, false, false);
}

template<int EPI>
__global__ __launch_bounds__(128)
void gemm_wmma(const float* __restrict__ A, int lda,
               const float* __restrict__ W,
               float* __restrict__ C, int ldc,
               int N, int K,
               const float* __restrict__ bias,
               const float* __restrict__ res, int ldr)
{
    __shared__ float ldsA[2][16 * LDSA_STRIDE];
    const int tid  = threadIdx.x;
    const int lane = tid & 31;
    const int wv   = tid >> 5;
    const int l15  = lane & 15;
    const int lh   = lane >> 4;            // K-pair selector for f32 frags
    const int m0   = blockIdx.x * 16;
    const int wn0  = blockIdx.y * 256 + wv * 64;
    const bool active = (wn0 < N);
    const int wn0c = active ? wn0 : 0;     // clamp so addresses stay valid

    // unconditional pointer setup -> stays in global addrspace
    const float* wp0 = W + (size_t)(wn0c +  0 + l15) * K + lh*2;
    const float* wp1 = W + (size_t)(wn0c + 16 + l15) * K + lh*2;
    const float* wp2 = W + (size_t)(wn0c + 32 + l15) * K + lh*2;
    const float* wp3 = W + (size_t)(wn0c + 48 + l15) * K + lh*2;

    v8f acc0 = {}, acc1 = {}, acc2 = {}, acc3 = {};

    // cooperative 16x32 A-chunk staging (one float4 per thread)
    const int arow = tid >> 3;             // 0..15
    const int acol = (tid & 7) * 4;        // 0..28
    const float* agp = A + (size_t)(m0 + arow) * lda + acol;
    float* alds0 = &ldsA[0][arow * LDSA_STRIDE + acol];
    float* alds1 = &ldsA[1][arow * LDSA_STRIDE + acol];
    const float* af0 = &ldsA[0][l15 * LDSA_STRIDE + lh*2];
    const float* af1 = &ldsA[1][l15 * LDSA_STRIDE + lh*2];

    const int nchunks = K / KC;
    {   // prologue: stage chunk 0
        const float4 g = *(const float4*)agp;
        alds0[0] = g.x; alds0[1] = g.y; alds0[2] = g.z; alds0[3] = g.w;
    }
    __syncthreads();

    for (int c = 0; c < nchunks; ++c) {
        const int kb = c * KC;
        const bool more = (c + 1 < nchunks);
        float4 gnext = {};
        if (more) gnext = *(const float4*)(agp + kb + KC);   // prefetch next A chunk

        if (active) {
            v2f bf0[8], bf1[8], bf2[8], bf3[8];
#pragma unroll
            for (int j = 0; j < 8; ++j) {
                bf0[j] = *(const v2f*)(wp0 + kb + j*4);
                bf1[j] = *(const v2f*)(wp1 + kb + j*4);
                bf2[j] = *(const v2f*)(wp2 + kb + j*4);
                bf3[j] = *(const v2f*)(wp3 + kb + j*4);
            }
            const float* af = (c & 1) ? af1 : af0;
#pragma unroll
            for (int j = 0; j < 8; ++j) {
                const v2f a = *(const v2f*)(af + j*4);
                acc0 = wmma4(a, bf0[j], acc0);
                acc1 = wmma4(a, bf1[j], acc1);
                acc2 = wmma4(a, bf2[j], acc2);
                acc3 = wmma4(a, bf3[j], acc3);
            }
        }
        if (more) {   // store prefetched chunk into the other LDS buffer
            float* alds = ((c + 1) & 1) ? alds1 : alds0;
            alds[0] = gnext.x; alds[1] = gnext.y; alds[2] = gnext.z; alds[3] = gnext.w;
        }
        __syncthreads();
    }
    if (!active) return;

#pragma unroll
    for (int nt = 0; nt < 4; ++nt) {
        const int n = wn0 + nt*16 + l15;
        const v8f acc = (nt == 0) ? acc0 : (nt == 1) ? acc1 : (nt == 2) ? acc2 : acc3;
        const float bia = (EPI == 1) ? bias[n] : 0.f;
#pragma unroll
        for (int r = 0; r < 8; ++r) {
            const size_t m = (size_t)(m0 + lh*8 + r);
            float v = acc[r];
            if (EPI == 1) { v += bia; v = (v > 20.f) ? v : log1pf(expf(v)); }
            else if (EPI == 2) { v += res[m * (size_t)ldr + n]; }
            C[m * (size_t)ldc + n] = v;
        }
    }
}

// ------------------------------------------------- depthwise causal conv + SiLU
__global__ __launch_bounds__(256)
void conv_silu_kernel(const float* __restrict__ xz,   // [M, 2048], xi in cols 0..1023
                      const float* __restrict__ cw,   // [1024, 4]
                      const float* __restrict__ cb,   // [1024]
                      float* __restrict__ xc)         // [M, 1024]
{
    const int idx = blockIdx.x * 256 + threadIdx.x;   // b*L*E + l*E + e
    const int e = idx & (D_INNER - 1);
    const int l = (idx >> 10) & (SEQ - 1);
    const int b = idx >> 21;
    float acc = cb[e];
#pragma unroll
    for (int j = 0; j < 4; ++j) {
        const int lj = l - 3 + j;
        if (lj >= 0)
            acc += cw[e*4 + j] * xz[(size_t)(b*SEQ + lj) * (2*D_INNER) + e];
    }
    acc = acc / (1.f + expf(-acc));                   // SiLU
    xc[(size_t)idx] = acc;
}

// ---------------------------------------------------------------- selective scan
// block = 256 thr = 16 channels x 16 states, one batch b; 512 blocks total.
// Writes gated output y into the dead xi half of xz (stride 2048).
__global__ __launch_bounds__(256)
void scan_kernel(const float* __restrict__ delta,  // [M, 1024]
                 const float* __restrict__ u,      // [M, 1024] (conv+silu)
                 const float* __restrict__ dbc,    // [M, 64]  (B cols 32..47, C 48..63)
                 const float* __restrict__ xz,     // [M, 2048] (z in cols 1024..2047)
                 const float* __restrict__ A_log,  // [1024, 16]
                 const float* __restrict__ Dp,     // [1024]
                 float* __restrict__ y)            // [M, 2048] (write cols 0..1023)
{
    const int T = 64;
    __shared__ float sB[T][16];
    __shared__ float sC[T][16];
    __shared__ float sDel[T][16];
    __shared__ float sU[T][16];
    __shared__ float sY[T][16];

    const int b  = blockIdx.x >> 6;      // 0..7
    const int eg = blockIdx.x & 63;      // channel group of 16
    const int tid = threadIdx.x;
    const int ch = tid >> 4;             // 0..15
    const int n  = tid & 15;             // state index
    const int e  = eg*16 + ch;

    const float An = -expf(A_log[e*D_STATE + n]);
    float h = 0.f;
    const size_t rowbase = (size_t)b * SEQ;

    for (int t0 = 0; t0 < SEQ; t0 += T) {
        __syncthreads();
        for (int i = tid; i < T*32; i += 256) {       // stage B,C
            const int t = i >> 5, c = i & 31;
            const float v = dbc[(rowbase + t0 + t)*64 + 32 + c];
            if (c < 16) sB[t][c] = v; else sC[t][c-16] = v;
        }
        for (int i = tid; i < T*16; i += 256) {       // stage delta,u
            const int t = i >> 4, c = i & 15;
            sDel[t][c] = delta[(rowbase + t0 + t)*D_INNER + eg*16 + c];
            sU[t][c]   = u    [(rowbase + t0 + t)*D_INNER + eg*16 + c];
        }
        __syncthreads();
        for (int t = 0; t < T; ++t) {
            const float d  = sDel[t][ch];
            const float uu = sU[t][ch];
            const float dA = expf(d * An);
            h = dA * h + (d * uu) * sB[t][n];
            float p = h * sC[t][n];
            p += __shfl_xor(p, 1, 32);
            p += __shfl_xor(p, 2, 32);
            p += __shfl_xor(p, 4, 32);
            p += __shfl_xor(p, 8, 32);
            if (n == 0) sY[t][ch] = p;
        }
        __syncthreads();
        for (int i = tid; i < T*16; i += 256) {       // gated, coalesced store
            const int t = i >> 4, c = i & 15;
            const size_t row = rowbase + t0 + t;
            const float z = xz[row*(2*D_INNER) + D_INNER + eg*16 + c];
            const float val = (sY[t][c] + sU[t][c]*Dp[eg*16 + c])
                              * (z / (1.f + expf(-z)));
            y[row*(2*D_INNER) + eg*16 + c] = val;
        }
    }
}

// ---------------------------------------------------------------- launcher
extern "C" void kernel_launch(void* const* d_in, const int* in_sizes, int n_in,
                              void* d_out, int out_size, void* d_ws, size_t ws_size,
                              hipStream_t stream)
{
    (void)in_sizes; (void)n_in; (void)out_size; (void)ws_size;
    const float* x      = (const float*)d_in[0];
    const float* ln_w   = (const float*)d_in[1];
    const float* ln_b   = (const float*)d_in[2];
    const float* in_w   = (const float*)d_in[3];
    const float* conv_w = (const float*)d_in[4];
    const float* conv_b = (const float*)d_in[5];
    const float* xp_w   = (const float*)d_in[6];
    const float* dt_w   = (const float*)d_in[7];
    const float* dt_b   = (const float*)d_in[8];
    const float* A_log  = (const float*)d_in[9];
    const float* Dp     = (const float*)d_in[10];
    const float* out_w  = (const float*)d_in[11];
    float* out = (float*)d_out;

    const size_t M = M_ROWS;
    float* ws    = (float*)d_ws;
    float* xn    = ws;                      // [M, 512]
    float* xz    = xn    + M * D_MODEL;     // [M, 2048]
    float* xc    = xz    + M * 2*D_INNER;   // [M, 1024]
    float* dbc   = xc    + M * D_INNER;     // [M, 64]
    float* delta = dbc   + M * 64;          // [M, 1024]
    float* x1    = delta + M * D_INNER;     // [M, 512] inter-layer residual

    const float* xin = x;
    for (int i = 0; i < 2; ++i) {
        float* xout = (i == 0) ? x1 : out;

        ln_kernel<<<M_ROWS/8, 256, 0, stream>>>(
            xin, ln_w + i*D_MODEL, ln_b + i*D_MODEL, xn);

        gemm_wmma<0><<<dim3(M_ROWS/16, (2*D_INNER)/256), 128, 0, stream>>>(
            xn, D_MODEL, in_w + (size_t)i*2*D_INNER*D_MODEL,
            xz, 2*D_INNER, 2*D_INNER, D_MODEL, nullptr, nullptr, 0);

        conv_silu_kernel<<<(M_ROWS*D_INNER)/256, 256, 0, stream>>>(
            xz, conv_w + i*D_INNER*4, conv_b + i*D_INNER, xc);

        gemm_wmma<0><<<dim3(M_ROWS/16, 1), 128, 0, stream>>>(
            xc, D_INNER, xp_w + (size_t)i*64*D_INNER,
            dbc, 64, 64, D_INNER, nullptr, nullptr, 0);

        gemm_wmma<1><<<dim3(M_ROWS/16, D_INNER/256), 128, 0, stream>>>(
            dbc, 64, dt_w + (size_t)i*D_INNER*DT_RANK,
            delta, D_INNER, D_INNER, DT_RANK, dt_b + i*D_INNER, nullptr, 0);

        scan_kernel<<<B_SZ*(D_INNER/16), 256, 0, stream>>>(
            delta, xc, dbc, xz,
            A_log + (size_t)i*D_INNER*D_STATE, Dp + i*D_INNER, xz);

        gemm_wmma<2><<<dim3(M_ROWS/16, D_MODEL/256), 128, 0, stream>>>(
            xz, 2*D_INNER, out_w + (size_t)i*D_MODEL*D_INNER,
            xout, D_MODEL, D_MODEL, D_INNER, nullptr, xin, D_MODEL);

        xin = xout;
    }
}